// GraphWaveNet_5463198400725
// MI455X (gfx1250) — compile-verified
//
#include <hip/hip_runtime.h>
#include <math.h>

// ---------------------------------------------------------------------------
// GraphWaveNet forward for MI455X (gfx1250, wave32, WMMA f32 16x16x4).
// All dense GEMMs use V_WMMA_F32_16X16X4_F32; SpMM is CSR-based, atomic-free
// and deterministic. skip/end MLP evaluated only at t = T-1 (output slice).
// ---------------------------------------------------------------------------

typedef __attribute__((ext_vector_type(2))) float v2f;
typedef __attribute__((ext_vector_type(8))) float v8f;

#define WMMA4(acc, a, b)                                                      \
  acc = __builtin_amdgcn_wmma_f32_16x16x4_f32(false, (a), false, (b),         \
                                              (short)0, (acc), false, false)

constexpr int Bn = 8, Tn = 32, Nn = 1024, En = 8192, Ln = 8, Kn = 2;
constexpr int CRn = 32, CDn = 32, CSn = 128, CEn = 256, PREDn = 12, EMBn = 16;
constexpr int NT = Nn * Tn;                 // 32768 (channel stride)
constexpr size_t ACT = (size_t)Bn * CRn * NT; // 8,388,608 floats per tensor

// ------------------------------- utility -----------------------------------

__global__ void zero_kernel(float* __restrict__ p, int n) {
  int i = blockIdx.x * blockDim.x + threadIdx.x;
  if (i < n) p[i] = 0.0f;
}

// h[b,c,n,t] = start_w[c,0]*x[b,t,n,0] + start_w[c,1]*x[b,t,n,1] + start_b[c]
__global__ void start_kernel(const float* __restrict__ x,
                             const float* __restrict__ sw,
                             const float* __restrict__ sb,
                             float* __restrict__ h) {
  int gid = blockIdx.x * blockDim.x + threadIdx.x;   // == flat h index
  int t = gid & 31;
  int n = (gid >> 5) & (Nn - 1);
  int c = (gid >> 15) & 31;
  int b = gid >> 20;
  const float* xp = x + (((size_t)b * Tn + t) * Nn + n) * 2;
  h[gid] = sw[c * 2] * xp[0] + sw[c * 2 + 1] * xp[1] + sb[c];
}

// -------------------------- graph preprocessing -----------------------------

__global__ void score_kernel(const int* __restrict__ bsrc,
                             const int* __restrict__ bdst,
                             const float* __restrict__ e1,
                             const float* __restrict__ e2,
                             float* __restrict__ se) {
  int e = blockIdx.x * blockDim.x + threadIdx.x;
  if (e >= En) return;
  int s = bsrc[e], d = bdst[e];
  float acc = 0.0f;
#pragma unroll
  for (int k = 0; k < EMBn; ++k) acc += e1[s * EMBn + k] * e2[d * EMBn + k];
  se[e] = 1.0f / (1.0f + __expf(-acc));
}

__global__ void count_kernel(const int* __restrict__ src, int* __restrict__ cnt) {
  int e = blockIdx.x * blockDim.x + threadIdx.x;
  if (e < En) atomicAdd(&cnt[src[e]], 1);          // int atomics: deterministic
}

__global__ void scan_kernel(const int* __restrict__ cnt, int* __restrict__ rp) {
  __shared__ int sc[Nn];
  int i = threadIdx.x;
  int v0 = cnt[i];
  sc[i] = v0;
  __syncthreads();
  for (int off = 1; off < Nn; off <<= 1) {
    int v = (i >= off) ? sc[i - off] : 0;
    __syncthreads();
    sc[i] += v;
    __syncthreads();
  }
  rp[i] = sc[i] - v0;                               // exclusive prefix
  if (i == Nn - 1) rp[Nn] = sc[i];
}

// Deterministic CSR fill: rank of edge e within its src row = #earlier edges
// with the same src.  O(E^2) total but one-time and tiny (33M cached reads).
__global__ void csr_fill_kernel(const int* __restrict__ src,
                                const int* __restrict__ dst,
                                const float* __restrict__ w_in,  // may be null
                                const int* __restrict__ rp,
                                int* __restrict__ cdst,
                                float* __restrict__ cw,
                                int* __restrict__ eidx) {
  int e = blockIdx.x * blockDim.x + threadIdx.x;
  if (e >= En) return;
  int s = src[e];
  int rank = 0;
  for (int q = 0; q < e; ++q) rank += (src[q] == s) ? 1 : 0;
  int pos = rp[s] + rank;
  cdst[pos] = dst[e];
  eidx[pos] = e;
  if (w_in) cw[pos] = w_in[e];
}

// Row-normalize sigmoid scores for the adaptive graph (deterministic order).
__global__ void base_norm_kernel(const int* __restrict__ rp,
                                 const int* __restrict__ eidx,
                                 const float* __restrict__ se,
                                 float* __restrict__ cw) {
  int i = blockIdx.x * blockDim.x + threadIdx.x;
  if (i >= Nn) return;
  int e0 = rp[i], e1 = rp[i + 1];
  float sum = 0.0f;
  for (int p = e0; p < e1; ++p) sum += se[eidx[p]];
  float den = sum + 1e-8f;
  for (int p = e0; p < e1; ++p) cw[p] = se[eidx[p]] / den;
}

// ------------------------- dilated conv + gate (WMMA) -----------------------
// xt[b,o,n,t] = tanh(f) * sigmoid(g);  f = sum_c fw[o,c,0]*h[t-d] + fw[o,c,1]*h[t]
__global__ void dconv_kernel(const float* __restrict__ h,
                             const float* __restrict__ fw,
                             const float* __restrict__ fb,
                             const float* __restrict__ gw,
                             const float* __restrict__ gb,
                             float* __restrict__ xt, int l, int d) {
  int gid = blockIdx.x * blockDim.x + threadIdx.x;
  int wave = gid >> 5, lane = gid & 31;
  int thalf = wave & 1;
  int n = (wave >> 1) & (Nn - 1);
  int b = wave >> 11;
  int col = lane & 15, khalf = lane >> 4;
  int t = thalf * 16 + col;
  const float* hc = h + (size_t)b * CRn * NT + (size_t)n * Tn + t;

  // B-fragments: current tap (colT) and delayed tap (colD), per 4-channel chunk
  v2f colT[8], colD[8];
  bool ok = (t - d) >= 0;
#pragma unroll
  for (int kc = 0; kc < 8; ++kc) {
    int c0 = 4 * kc + 2 * khalf;
    colT[kc][0] = hc[(size_t)c0 * NT];
    colT[kc][1] = hc[(size_t)(c0 + 1) * NT];
    colD[kc][0] = ok ? hc[(size_t)c0 * NT - d] : 0.0f;
    colD[kc][1] = ok ? hc[(size_t)(c0 + 1) * NT - d] : 0.0f;
  }

  v8f accF0 = {}, accF1 = {}, accG0 = {}, accG1 = {};
#pragma unroll
  for (int kc = 0; kc < 8; ++kc) {
    int c0 = 4 * kc + 2 * khalf;
    {  // out channels 0..15
      const float* fp = fw + (((size_t)l * CDn + col) * CRn + c0) * Kn;
      const float* gp = gw + (((size_t)l * CDn + col) * CRn + c0) * Kn;
      v2f aF0 = {fp[0], fp[2]}, aF1 = {fp[1], fp[3]};
      v2f aG0 = {gp[0], gp[2]}, aG1 = {gp[1], gp[3]};
      WMMA4(accF0, aF0, colD[kc]);
      WMMA4(accF0, aF1, colT[kc]);
      WMMA4(accG0, aG0, colD[kc]);
      WMMA4(accG0, aG1, colT[kc]);
    }
    {  // out channels 16..31
      const float* fp = fw + (((size_t)l * CDn + 16 + col) * CRn + c0) * Kn;
      const float* gp = gw + (((size_t)l * CDn + 16 + col) * CRn + c0) * Kn;
      v2f aF0 = {fp[0], fp[2]}, aF1 = {fp[1], fp[3]};
      v2f aG0 = {gp[0], gp[2]}, aG1 = {gp[1], gp[3]};
      WMMA4(accF1, aF0, colD[kc]);
      WMMA4(accF1, aF1, colT[kc]);
      WMMA4(accG1, aG0, colD[kc]);
      WMMA4(accG1, aG1, colT[kc]);
    }
  }

  float* xc = xt + (size_t)b * CDn * NT + (size_t)n * Tn + t;
#pragma unroll
  for (int r = 0; r < 8; ++r) {
    int m = r + 8 * khalf;
    {
      float f = accF0[r] + fb[l * CDn + m];
      float g = accG0[r] + gb[l * CDn + m];
      xc[(size_t)m * NT] = tanhf(f) * (1.0f / (1.0f + __expf(-g)));
    }
    {
      int c = 16 + m;
      float f = accF1[r] + fb[l * CDn + c];
      float g = accG1[r] + gb[l * CDn + c];
      xc[(size_t)c * NT] = tanhf(f) * (1.0f / (1.0f + __expf(-g)));
    }
  }
}

// ------------- skip accumulation (only t = T-1 survives to output) ----------
__global__ void skip_kernel(const float* __restrict__ xt,
                            const float* __restrict__ sw,
                            const float* __restrict__ sb,
                            float* __restrict__ sl, int l) {
  int gid = blockIdx.x * blockDim.x + threadIdx.x;
  int wave = gid >> 5, lane = gid & 31;
  int b = wave >> 6;
  int n0 = (wave & 63) * 16;
  int col = lane & 15, khalf = lane >> 4;
  int n = n0 + col;

  v2f bcol[8];
#pragma unroll
  for (int kc = 0; kc < 8; ++kc) {
    int c0 = 4 * kc + 2 * khalf;
    bcol[kc][0] = xt[((size_t)b * CDn + c0) * NT + (size_t)n * Tn + (Tn - 1)];
    bcol[kc][1] = xt[((size_t)b * CDn + c0 + 1) * NT + (size_t)n * Tn + (Tn - 1)];
  }

  v8f acc[8];
#pragma unroll
  for (int ot = 0; ot < 8; ++ot)
#pragma unroll
    for (int r = 0; r < 8; ++r) {
      int o = ot * 16 + r + 8 * khalf;
      acc[ot][r] = sl[((size_t)b * CSn + o) * Nn + n];
    }

#pragma unroll
  for (int ot = 0; ot < 8; ++ot)
#pragma unroll
    for (int kc = 0; kc < 8; ++kc) {
      int c0 = 4 * kc + 2 * khalf;
      const float* wp = sw + ((size_t)l * CSn + ot * 16 + col) * CDn + c0;
      v2f a = {wp[0], wp[1]};
      WMMA4(acc[ot], a, bcol[kc]);
    }

#pragma unroll
  for (int ot = 0; ot < 8; ++ot)
#pragma unroll
    for (int r = 0; r < 8; ++r) {
      int o = ot * 16 + r + 8 * khalf;
      sl[((size_t)b * CSn + o) * Nn + n] = acc[ot][r] + sb[l * CSn + o];
    }
}

// ------------------------------- SpMM ---------------------------------------
// out[b,:,i,:] = sum_{edges of row i} w * x[b,:,dst,:]   (one wave per (b,i),
// lane = t; coalesced 128B row gathers, register accumulation, no atomics)
__global__ void spmm_kernel(const int* __restrict__ rp,
                            const int* __restrict__ cdst,
                            const float* __restrict__ cw,
                            const float* __restrict__ x,
                            float* __restrict__ out) {
  int gid = blockIdx.x * blockDim.x + threadIdx.x;
  int wave = gid >> 5, t = gid & 31;
  int b = wave >> 10;
  int i = wave & (Nn - 1);
  int e0 = rp[i], e1 = rp[i + 1];
  float acc[32];
#pragma unroll
  for (int c = 0; c < 32; ++c) acc[c] = 0.0f;
  size_t bbase = (size_t)b * CDn * NT;
  for (int e = e0; e < e1; ++e) {
    int dj = cdst[e];
    float w = cw[e];
    const float* xp = x + bbase + (size_t)dj * Tn + t;
#pragma unroll
    for (int c = 0; c < 32; ++c) acc[c] += w * xp[(size_t)c * NT];
  }
  float* op = out + bbase + (size_t)i * Tn + t;
#pragma unroll
  for (int c = 0; c < 32; ++c) op[(size_t)c * NT] = acc[c];
}

// --------- fused gconv(32x160) -> res(32x32) -> +residual -> BN -------------
// Stage-1 WMMA result is staged through LDS (bank-padded) to become the
// stage-2 B-fragments; epilogue applies res_b, residual add and BatchNorm.
__global__ void gcfused_kernel(const float* __restrict__ xt,
                               const float* __restrict__ p1,
                               const float* __restrict__ p2,
                               const float* __restrict__ p3,
                               const float* __restrict__ p4,
                               const float* __restrict__ gw,
                               const float* __restrict__ gcb,
                               const float* __restrict__ rw,
                               const float* __restrict__ rb,
                               const float* __restrict__ bng,
                               const float* __restrict__ bnb,
                               const float* __restrict__ bnm,
                               const float* __restrict__ bnv,
                               float* __restrict__ h, int l) {
  __shared__ float stage[8][32 * 17];  // per-wave 32x16 tile, pad 17 vs banks
  int gid = blockIdx.x * blockDim.x + threadIdx.x;
  int wave = gid >> 5, lane = gid & 31;
  int wib = threadIdx.x >> 5;
  int thalf = wave & 1;
  int n = (wave >> 1) & (Nn - 1);
  int b = wave >> 11;
  int col = lane & 15, khalf = lane >> 4;
  int t = thalf * 16 + col;
  size_t pbase = (size_t)b * CDn * NT + (size_t)n * Tn + t;

  const float* parts[5] = {xt, p1, p2, p3, p4};
  v8f acc0 = {}, acc1 = {};
#pragma unroll
  for (int j = 0; j < 5; ++j) {
    const float* part = parts[j];
#pragma unroll
    for (int kc = 0; kc < 8; ++kc) {
      int c0 = 4 * kc + 2 * khalf;
      v2f bfr = {part[pbase + (size_t)c0 * NT], part[pbase + (size_t)(c0 + 1) * NT]};
      const float* w0 = gw + ((size_t)l * CRn + col) * (5 * CDn) + j * CDn + c0;
      const float* w1 = gw + ((size_t)l * CRn + 16 + col) * (5 * CDn) + j * CDn + c0;
      v2f a0 = {w0[0], w0[1]};
      v2f a1 = {w1[0], w1[1]};
      WMMA4(acc0, a0, bfr);
      WMMA4(acc1, a1, bfr);
    }
  }

  float* sl = &stage[wib][0];
#pragma unroll
  for (int r = 0; r < 8; ++r) {
    int m = r + 8 * khalf;
    sl[m * 17 + col] = acc0[r] + gcb[l * CRn + m];
    sl[(16 + m) * 17 + col] = acc1[r] + gcb[l * CRn + 16 + m];
  }
  __syncthreads();

  v8f racc0 = {}, racc1 = {};
#pragma unroll
  for (int kc = 0; kc < 8; ++kc) {
    int c0 = 4 * kc + 2 * khalf;
    v2f bfr = {sl[c0 * 17 + col], sl[(c0 + 1) * 17 + col]};
    const float* w0 = rw + ((size_t)l * CRn + col) * CDn + c0;
    const float* w1 = rw + ((size_t)l * CRn + 16 + col) * CDn + c0;
    v2f a0 = {w0[0], w0[1]};
    v2f a1 = {w1[0], w1[1]};
    WMMA4(racc0, a0, bfr);
    WMMA4(racc1, a1, bfr);
  }

  float* hc = h + (size_t)b * CRn * NT + (size_t)n * Tn + t;
#pragma unroll
  for (int r = 0; r < 8; ++r) {
    int m = r + 8 * khalf;
    {
      float inv = bng[l * CRn + m] * rsqrtf(bnv[l * CRn + m] + 1e-5f);
      float v = racc0[r] + rb[l * CRn + m] + hc[(size_t)m * NT];
      hc[(size_t)m * NT] = v * inv + (bnb[l * CRn + m] - bnm[l * CRn + m] * inv);
    }
    {
      int c = 16 + m;
      float inv = bng[l * CRn + c] * rsqrtf(bnv[l * CRn + c] + 1e-5f);
      float v = racc1[r] + rb[l * CRn + c] + hc[(size_t)c * NT];
      hc[(size_t)c * NT] = v * inv + (bnb[l * CRn + c] - bnm[l * CRn + c] * inv);
    }
  }
}

// --------- final MLP at t=T-1: relu -> end1(256x128) -> relu -> end2(12x256)
__global__ void end_kernel(const float* __restrict__ sl,
                           const float* __restrict__ e1w,
                           const float* __restrict__ e1b,
                           const float* __restrict__ e2w,
                           const float* __restrict__ e2b,
                           float* __restrict__ out) {
  __shared__ float y1[CEn * 17];
  int lane = threadIdx.x;
  int tile = blockIdx.x;
  int b = tile >> 6;
  int n0 = (tile & 63) * 16;
  int col = lane & 15, khalf = lane >> 4;
  int n = n0 + col;

  v2f bcol[32];  // relu(skip) column, 128 channels
#pragma unroll
  for (int kc = 0; kc < 32; ++kc) {
    int c0 = 4 * kc + 2 * khalf;
    bcol[kc][0] = fmaxf(sl[((size_t)b * CSn + c0) * Nn + n], 0.0f);
    bcol[kc][1] = fmaxf(sl[((size_t)b * CSn + c0 + 1) * Nn + n], 0.0f);
  }

  for (int ot = 0; ot < 16; ++ot) {  // 256 output channels of end1
    v8f acc = {};
#pragma unroll
    for (int kc = 0; kc < 32; ++kc) {
      int c0 = 4 * kc + 2 * khalf;
      const float* wp = e1w + (size_t)(ot * 16 + col) * CSn + c0;
      v2f a = {wp[0], wp[1]};
      WMMA4(acc, a, bcol[kc]);
    }
#pragma unroll
    for (int r = 0; r < 8; ++r) {
      int o = ot * 16 + r + 8 * khalf;
      y1[o * 17 + col] = fmaxf(acc[r] + e1b[o], 0.0f);
    }
  }
  __syncthreads();

  v8f acc2 = {};
  int oa = (col < PREDn) ? col : (PREDn - 1);  // clamp A rows (pad 12 -> 16)
#pragma unroll
  for (int kc = 0; kc < 64; ++kc) {
    int c0 = 4 * kc + 2 * khalf;
    v2f bfr = {y1[c0 * 17 + col], y1[(c0 + 1) * 17 + col]};
    const float* wp = e2w + (size_t)oa * CEn + c0;
    v2f a = {wp[0], wp[1]};
    WMMA4(acc2, a, bfr);
  }
#pragma unroll
  for (int r = 0; r < 8; ++r) {
    int m = r + 8 * khalf;
    if (m < PREDn) out[((size_t)b * PREDn + m) * Nn + n] = acc2[r] + e2b[m];
  }
}

// ------------------------------- launcher -----------------------------------

static inline void* carve(char*& p, size_t bytes) {
  void* r = p;
  p += (bytes + 255) & ~(size_t)255;
  return r;
}

extern "C" void kernel_launch(void* const* d_in, const int* in_sizes, int n_in,
                              void* d_out, int out_size, void* d_ws,
                              size_t ws_size, hipStream_t stream) {
  (void)in_sizes; (void)n_in; (void)out_size; (void)ws_size;
  const float* x        = (const float*)d_in[0];
  const int*   sup_src  = (const int*)d_in[1];
  const int*   sup_dst  = (const int*)d_in[2];
  const float* sup_w    = (const float*)d_in[3];
  const int*   base_src = (const int*)d_in[4];
  const int*   base_dst = (const int*)d_in[5];
  const float* emb1     = (const float*)d_in[6];
  const float* emb2     = (const float*)d_in[7];
  const float* start_w  = (const float*)d_in[8];
  const float* start_b  = (const float*)d_in[9];
  const float* filt_w   = (const float*)d_in[10];
  const float* filt_b   = (const float*)d_in[11];
  const float* gate_w   = (const float*)d_in[12];
  const float* gate_b   = (const float*)d_in[13];
  const float* res_w    = (const float*)d_in[14];
  const float* res_b    = (const float*)d_in[15];
  const float* skip_w   = (const float*)d_in[16];
  const float* skip_b   = (const float*)d_in[17];
  const float* gconv_w  = (const float*)d_in[18];
  const float* gconv_b  = (const float*)d_in[19];
  const float* bn_gamma = (const float*)d_in[20];
  const float* bn_beta  = (const float*)d_in[21];
  const float* bn_mean  = (const float*)d_in[22];
  const float* bn_var   = (const float*)d_in[23];
  const float* end1_w   = (const float*)d_in[24];
  const float* end1_b   = (const float*)d_in[25];
  const float* end2_w   = (const float*)d_in[26];
  const float* end2_b   = (const float*)d_in[27];

  // workspace layout (~206 MB): 6 activation buffers + skip_last + graph prep
  char* wsp = (char*)d_ws;
  float* H   = (float*)carve(wsp, ACT * sizeof(float));
  float* XT  = (float*)carve(wsp, ACT * sizeof(float));
  float* P1  = (float*)carve(wsp, ACT * sizeof(float));
  float* P2  = (float*)carve(wsp, ACT * sizeof(float));
  float* P3  = (float*)carve(wsp, ACT * sizeof(float));
  float* P4  = (float*)carve(wsp, ACT * sizeof(float));
  float* SL  = (float*)carve(wsp, (size_t)Bn * CSn * Nn * sizeof(float));
  float* SE  = (float*)carve(wsp, En * sizeof(float));
  int*   cntS  = (int*)carve(wsp, Nn * sizeof(int));
  int*   cntB  = (int*)carve(wsp, Nn * sizeof(int));
  int*   rpS   = (int*)carve(wsp, (Nn + 1) * sizeof(int));
  int*   rpB   = (int*)carve(wsp, (Nn + 1) * sizeof(int));
  int*   cdstS = (int*)carve(wsp, En * sizeof(int));
  int*   cdstB = (int*)carve(wsp, En * sizeof(int));
  float* cwS   = (float*)carve(wsp, En * sizeof(float));
  float* cwB   = (float*)carve(wsp, En * sizeof(float));
  int*   eiS   = (int*)carve(wsp, En * sizeof(int));
  int*   eiB   = (int*)carve(wsp, En * sizeof(int));

  // --- init accumulators / counts ---
  zero_kernel<<<4096, 256, 0, stream>>>(SL, Bn * CSn * Nn);
  zero_kernel<<<4, 256, 0, stream>>>((float*)cntS, Nn);
  zero_kernel<<<4, 256, 0, stream>>>((float*)cntB, Nn);

  // --- start conv ---
  start_kernel<<<32768, 256, 0, stream>>>(x, start_w, start_b, H);

  // --- graph preprocessing (deterministic, no float atomics) ---
  score_kernel<<<32, 256, 0, stream>>>(base_src, base_dst, emb1, emb2, SE);
  count_kernel<<<32, 256, 0, stream>>>(sup_src, cntS);
  count_kernel<<<32, 256, 0, stream>>>(base_src, cntB);
  scan_kernel<<<1, 1024, 0, stream>>>(cntS, rpS);
  scan_kernel<<<1, 1024, 0, stream>>>(cntB, rpB);
  csr_fill_kernel<<<32, 256, 0, stream>>>(sup_src, sup_dst, sup_w, rpS, cdstS,
                                          cwS, eiS);
  csr_fill_kernel<<<32, 256, 0, stream>>>(base_src, base_dst, nullptr, rpB,
                                          cdstB, cwB, eiB);
  base_norm_kernel<<<4, 256, 0, stream>>>(rpB, eiB, SE, cwB);

  // --- 8 WaveNet layers ---
  for (int l = 0; l < Ln; ++l) {
    int d = 1 << (l & 1);  // dilation 1,2,1,2,...
    dconv_kernel<<<2048, 256, 0, stream>>>(H, filt_w, filt_b, gate_w, gate_b,
                                           XT, l, d);
    skip_kernel<<<64, 256, 0, stream>>>(XT, skip_w, skip_b, SL, l);
    spmm_kernel<<<1024, 256, 0, stream>>>(rpS, cdstS, cwS, XT, P1);
    spmm_kernel<<<1024, 256, 0, stream>>>(rpS, cdstS, cwS, P1, P2);
    spmm_kernel<<<1024, 256, 0, stream>>>(rpB, cdstB, cwB, XT, P3);
    spmm_kernel<<<1024, 256, 0, stream>>>(rpB, cdstB, cwB, P3, P4);
    gcfused_kernel<<<2048, 256, 0, stream>>>(XT, P1, P2, P3, P4, gconv_w,
                                             gconv_b, res_w, res_b, bn_gamma,
                                             bn_beta, bn_mean, bn_var, H, l);
  }

  // --- output head (t = T-1 only) ---
  end_kernel<<<512, 32, 0, stream>>>(SL, end1_w, end1_b, end2_w, end2_b,
                                     (float*)d_out);
}